// QuanvolutionHybrid_65481071406712
// MI455X (gfx1250) — compile-verified
//
#include <hip/hip_runtime.h>
#include <math.h>

// ---------------------------------------------------------------------------
// QuanvolutionHybrid on gfx1250 (MI455X), restructured as f32 WMMA GEMMs.
//
//   psi = U * v       (U = fixed 16x16 variational unitary, v = real product
//                      state from encoder angles)  -> V_WMMA_F32_16X16X4_F32
//   ez  = WHT(|psi|^2) picked at single-bit masks  -> DS_SWIZZLE_B32 (xor)
//   logits = y @ fc_w^T + b ; log_softmax          -> V_WMMA_F32_16X16X4_F32
// ---------------------------------------------------------------------------

typedef float v2f __attribute__((ext_vector_type(2)));
typedef float v8f __attribute__((ext_vector_type(8)));

#define WMMA4(A, B, C) \
    __builtin_amdgcn_wmma_f32_16x16x4_f32(false, (A), false, (B), (short)0, (C), false, false)

// DS_SWIZZLE_B32, group-of-32 mode: new_lane = (lane & 0x1F) ^ MASK.
// For MASK < 16 this never crosses a 16-lane half, so it is exactly a
// width-16 xor shuffle without any clamp logic.
template <int MASK>
__device__ inline float xor_swz(float x) {
    return __int_as_float(
        __builtin_amdgcn_ds_swizzle(__float_as_int(x), (MASK << 10) | 0x1F));
}

// ------------------------- Kernel 1: build U -------------------------------
// Lane j (j<16) carries column j of U (the 16-amp complex state obtained by
// applying the variational circuit to basis state e_j). Templated gate
// helpers keep indices compile-time so the state lives in VGPRs.
// Runs once on a single wave -> keep precise trig here.

template <int BIT>
__device__ inline void g_ry(float (&ur)[16], float (&ui)[16], float th) {
    float c = cosf(th * 0.5f), s = sinf(th * 0.5f);
#pragma unroll
    for (int i = 0; i < 16; ++i) {
        if (!(i & BIT)) {
            const int k = i | BIT;
            float r0 = ur[i], r1 = ur[k], i0 = ui[i], i1 = ui[k];
            ur[i] = c * r0 - s * r1;  ur[k] = s * r0 + c * r1;
            ui[i] = c * i0 - s * i1;  ui[k] = s * i0 + c * i1;
        }
    }
}

template <int BIT>
__device__ inline void g_rz(float (&ur)[16], float (&ui)[16], float th) {
    float c = cosf(th * 0.5f), s = sinf(th * 0.5f);
#pragma unroll
    for (int i = 0; i < 16; ++i) {
        float r = ur[i], im = ui[i];
        if (i & BIT) {  // multiply by exp(+i th/2) = c + i s
            ur[i] = r * c - im * s;  ui[i] = im * c + r * s;
        } else {        // multiply by exp(-i th/2) = c - i s
            ur[i] = r * c + im * s;  ui[i] = im * c - r * s;
        }
    }
}

template <int CB, int TB>
__device__ inline void g_cnot(float (&ur)[16], float (&ui)[16]) {
#pragma unroll
    for (int i = 0; i < 16; ++i) {
        if ((i & CB) && !(i & TB)) {
            const int k = i | TB;
            float t = ur[i]; ur[i] = ur[k]; ur[k] = t;
            t = ui[i]; ui[i] = ui[k]; ui[k] = t;
        }
    }
}

__global__ void build_u_kernel(const float* __restrict__ vp, float* __restrict__ Uout) {
    const int lane = threadIdx.x & 31;
    const int j = lane & 15;
    float ur[16], ui[16];
#pragma unroll
    for (int i = 0; i < 16; ++i) { ur[i] = (i == j) ? 1.0f : 0.0f; ui[i] = 0.0f; }

    for (int l = 0; l < 3; ++l) {
        const float* p = vp + l * 8;  // [4 wires][2]
        g_ry<8>(ur, ui, p[0]); g_rz<8>(ur, ui, p[1]);   // wire 0 (bit 8)
        g_ry<4>(ur, ui, p[2]); g_rz<4>(ur, ui, p[3]);   // wire 1 (bit 4)
        g_ry<2>(ur, ui, p[4]); g_rz<2>(ur, ui, p[5]);   // wire 2 (bit 2)
        g_ry<1>(ur, ui, p[6]); g_rz<1>(ur, ui, p[7]);   // wire 3 (bit 1)
        g_cnot<8, 4>(ur, ui);
        g_cnot<4, 2>(ur, ui);
        g_cnot<2, 1>(ur, ui);
        g_cnot<1, 8>(ur, ui);
    }
    if (lane < 16) {
#pragma unroll
        for (int n = 0; n < 16; ++n) {
            Uout[n * 16 + j]       = ur[n];  // Re U, row-major [amp n][basis j]
            Uout[256 + n * 16 + j] = ui[n];  // Im U
        }
    }
}

// ------------------- Kernel 2: quanvolution via WMMA -----------------------
// One wave per tile of 16 patches. Psi(16x16, re+im) = V(16x16) @ U^T via
// 4+4 chained V_WMMA_F32_16X16X4_F32, then per-row Walsh-Hadamard over the
// 16 amplitude lanes (ds_swizzle xor butterflies) gives all four PauliZ
// expectations at lanes with n in {8,4,2,1}.
// Encoder angles are pixels in [0,1): theta/2 in [0,0.5) -> hardware
// v_sin/v_cos (__sinf/__cosf) need no range reduction.

__global__ void quanv_kernel(const float* __restrict__ x, const float* __restrict__ U,
                             float* __restrict__ y, int ntiles) {
    const int wave = (int)((blockIdx.x * blockDim.x + threadIdx.x) >> 5);
    const int lane = (int)(threadIdx.x & 31);
    if (wave >= ntiles) return;  // wave-uniform

    const int n    = lane & 15;      // B-matrix column / A-matrix row / D column
    const int half = lane >> 4;
    const int koff = half * 2;       // K sub-offset carried by upper lane half

    // Preload B = U^T chunks: B[K=4k+koff+t][N=n] = U[n][4k+koff+t]
    float bre[4][2], bim[4][2];
#pragma unroll
    for (int k = 0; k < 4; ++k) {
#pragma unroll
        for (int t = 0; t < 2; ++t) {
            bre[k][t] = U[n * 16 + 4 * k + koff + t];
            bim[k][t] = U[256 + n * 16 + 4 * k + koff + t];
        }
    }

    // A matrix: row m = n -> patch wave*16+n; both lane halves read the same
    // patch (they carry different K components of the same row).
    const int pg = wave * 16 + n;
    const int b  = pg / 196;
    const int p  = pg - b * 196;
    const int pr = p / 14;
    const int pc = p - pr * 14;
    const float* xb = x + (size_t)b * 784 + (2 * pr) * 28 + 2 * pc;
    const float a0 = xb[0], a1 = xb[1], a2 = xb[28], a3 = xb[29];

    const float c0 = __cosf(a0 * 0.5f), s0 = __sinf(a0 * 0.5f);
    const float c1 = __cosf(a1 * 0.5f), s1 = __sinf(a1 * 0.5f);
    const float c2 = __cosf(a2 * 0.5f), s2 = __sinf(a2 * 0.5f);
    const float c3 = __cosf(a3 * 0.5f), s3 = __sinf(a3 * 0.5f);

    // v[i] = prod_w (bit_w(i) ? sin : cos), wire0 -> bit 8.
    // chunk k covers i = 4k..4k+3 : (b0,b1) = bits of k ; b2 from lane half.
    const float w01[4] = {c0 * c1, c0 * s1, s0 * c1, s0 * s1};
    const float t2 = half ? s2 : c2;

    v2f a[4];
#pragma unroll
    for (int k = 0; k < 4; ++k) {
        const float w = w01[k] * t2;
        a[k].x = w * c3;   // K offset 0/2 -> b3 = 0
        a[k].y = w * s3;   // K offset 1/3 -> b3 = 1
    }

    v8f dre = {}, dim = {};
#pragma unroll
    for (int k = 0; k < 4; ++k) {
        v2f br; br.x = bre[k][0]; br.y = bre[k][1];
        v2f bi; bi.x = bim[k][0]; bi.y = bim[k][1];
        dre = WMMA4(a[k], br, dre);
        dim = WMMA4(a[k], bi, dim);
    }

    // One-hot lanes n in {8,4,2,1} own wire 0,1,2,3 respectively.
    const bool onehot = (n != 0) && ((n & (n - 1)) == 0);
    const int  wire   = (n == 8) ? 0 : (n == 4) ? 1 : (n == 2) ? 2 : 3;

    // D row r holds patch m = r + half*8 ; lane column n = amplitude index.
#pragma unroll
    for (int r = 0; r < 8; ++r) {
        const float pre = dre[r], pim = dim[r];
        float pz = pre * pre + pim * pim;
        // Walsh-Hadamard butterflies over the 16 amplitude lanes
        {
            float o = xor_swz<1>(pz); pz = (n & 1) ? (o - pz) : (pz + o);
            o = xor_swz<2>(pz);       pz = (n & 2) ? (o - pz) : (pz + o);
            o = xor_swz<4>(pz);       pz = (n & 4) ? (o - pz) : (pz + o);
            o = xor_swz<8>(pz);       pz = (n & 8) ? (o - pz) : (pz + o);
        }
        const int pgr = wave * 16 + r + half * 8;
        if (onehot) y[(size_t)pgr * 4 + wire] = pz;
    }
}

// ------------- Kernel 3: FC (784 -> 10) + log_softmax via WMMA -------------

__global__ void fc_kernel(const float* __restrict__ y, const float* __restrict__ fcw,
                          const float* __restrict__ fcb, float* __restrict__ out,
                          int ntiles) {
    const int wave = (int)((blockIdx.x * blockDim.x + threadIdx.x) >> 5);
    const int lane = (int)(threadIdx.x & 31);
    if (wave >= ntiles) return;  // wave-uniform

    const int n    = lane & 15;
    const int half = lane >> 4;
    const int koff = half * 2;

    const int  row = wave * 16 + n;           // batch row for this lane's A data
    const bool nv  = (n < 10);
    const int  nc  = nv ? n : 0;              // clamped class index (safe loads)

    const float* yrow = y + (size_t)row * 784;
    const float* wrow = fcw + (size_t)nc * 784;

    v8f acc = {};
    for (int k = 0; k < 196; ++k) {
        v2f av = *(const v2f*)(yrow + 4 * k + koff);
        v2f bv = *(const v2f*)(wrow + 4 * k + koff);
        bv.x = nv ? bv.x : 0.0f;
        bv.y = nv ? bv.y : 0.0f;
        acc = WMMA4(av, bv, acc);
    }

    const float bias = nv ? fcb[nc] : 0.0f;

#pragma unroll
    for (int r = 0; r < 8; ++r) {
        const float v = acc[r] + bias;
        // max over classes (xor butterflies stay inside each 16-lane row)
        float xm = nv ? v : -1e30f;
        xm = fmaxf(xm, xor_swz<8>(xm));
        xm = fmaxf(xm, xor_swz<4>(xm));
        xm = fmaxf(xm, xor_swz<2>(xm));
        xm = fmaxf(xm, xor_swz<1>(xm));
        float e = nv ? __expf(v - xm) : 0.0f;
        e += xor_swz<8>(e);
        e += xor_swz<4>(e);
        e += xor_swz<2>(e);
        e += xor_swz<1>(e);
        const float res = (v - xm) - __logf(e);
        const int orow = wave * 16 + r + half * 8;
        if (nv) out[(size_t)orow * 10 + n] = res;
    }
}

// ---------------------------------------------------------------------------

extern "C" void kernel_launch(void* const* d_in, const int* in_sizes, int n_in,
                              void* d_out, int out_size, void* d_ws, size_t ws_size,
                              hipStream_t stream) {
    const float* x   = (const float*)d_in[0];  // [B, 28, 28]
    const float* vp  = (const float*)d_in[1];  // [3, 4, 2]
    const float* fcw = (const float*)d_in[2];  // [10, 784]
    const float* fcb = (const float*)d_in[3];  // [10]
    float* out = (float*)d_out;                // [B, 10]

    const int B = in_sizes[0] / 784;

    float* U = (float*)d_ws;          // 512 floats (Re U | Im U)
    float* y = (float*)d_ws + 1024;   // [B, 784] expectations

    build_u_kernel<<<1, 32, 0, stream>>>(vp, U);

    const int npatch  = B * 196;
    const int qtiles  = npatch / 16;                 // 16 patches per wave
    const int qblocks = (qtiles * 32 + 255) / 256;   // 8 waves per block
    quanv_kernel<<<qblocks, 256, 0, stream>>>(x, U, y, qtiles);

    const int ftiles  = B / 16;                      // 16 batch rows per wave
    const int fblocks = (ftiles * 32 + 255) / 256;
    fc_kernel<<<fblocks, 256, 0, stream>>>(y, fcw, fcb, out, ftiles);
}